// Net_20212116095614
// MI455X (gfx1250) — compile-verified
//
#include <hip/hip_runtime.h>
#include <hip/hip_bf16.h>

// GCN (3x GCNConv + global_add_pool + linear) for MI455X / gfx1250.
// GEMMs use V_WMMA_F32_16X16X4_F32 (full-precision f32 matrix pipe).
// Edge aggregation is edge-parallel with native agent-scope f32 atomics.

typedef float v2f __attribute__((ext_vector_type(2)));
typedef float v8f __attribute__((ext_vector_type(8)));

#define DIMD 64
#define NNODES 100000
#define NGRAPHS 64

// ---------------- degree / normalization ----------------

__global__ __launch_bounds__(256) void k_init_deg(float* deg, int n) {
    int i = blockIdx.x * blockDim.x + threadIdx.x;
    if (i < n) deg[i] = 1.0f;  // self-loop contributes 1
}

__global__ __launch_bounds__(256) void k_deg_count(const int* __restrict__ dst,
                                                   float* __restrict__ deg, int nE) {
    int e = blockIdx.x * blockDim.x + threadIdx.x;
    if (e < nE)
        __hip_atomic_fetch_add(&deg[dst[e]], 1.0f, __ATOMIC_RELAXED,
                               __HIP_MEMORY_SCOPE_AGENT);
}

__global__ __launch_bounds__(256) void k_dis(const float* __restrict__ deg,
                                             float* __restrict__ dis,
                                             float* __restrict__ invdeg, int n) {
    int i = blockIdx.x * blockDim.x + threadIdx.x;
    if (i < n) {
        float r = rsqrtf(deg[i]);
        dis[i]    = r;       // D^{-1/2}
        invdeg[i] = r * r;   // self-loop norm dis[i]*dis[i]
    }
}

// ---------------- H = X @ W via WMMA f32 16x16x4 ----------------
// Block = 128 threads = 4 waves; wave w owns column tile c0 = 16*w of a
// 16-row slab. W (64x64 f32, 16KB) is staged in LDS.
// A-frag  (16x4):  lane<16 -> K={k,k+1}, lane>=16 -> K={k+2,k+3}, M = lane&15
// B-frag  (4x16):  same K split, N = lane&15
// C/D: VGPR i -> M = i + 8*(lane>=16), N = lane&15

__global__ __launch_bounds__(128) void k_gemm64(const float* __restrict__ X,
                                                const float* __restrict__ W,
                                                float* __restrict__ H) {
    __shared__ float sW[DIMD * DIMD];
    {
        const float4* Wv = (const float4*)W;
        float4* sWv = (float4*)sW;
        #pragma unroll
        for (int i = threadIdx.x; i < (DIMD * DIMD / 4); i += 128) sWv[i] = Wv[i];
    }
    __syncthreads();

    const int wave = threadIdx.x >> 5;
    const int lane = threadIdx.x & 31;
    const int half = lane >> 4;      // 0: K lo pair, 1: K hi pair
    const int l    = lane & 15;
    const int r0   = blockIdx.x * 16;      // 100000 % 16 == 0
    const int c0   = wave * 16;

    const float* __restrict__ xrow = X + (size_t)(r0 + l) * DIMD;

    v8f acc = {};
    #pragma unroll
    for (int k = 0; k < DIMD; k += 4) {
        const int kk = k + 2 * half;
        v2f a; a.x = xrow[kk];               a.y = xrow[kk + 1];
        v2f b; b.x = sW[kk * DIMD + c0 + l]; b.y = sW[(kk + 1) * DIMD + c0 + l];
        acc = __builtin_amdgcn_wmma_f32_16x16x4_f32(
                  false, a, false, b, (short)0, acc, false, false);
    }

    #pragma unroll
    for (int i = 0; i < 8; ++i)
        H[(size_t)(r0 + 8 * half + i) * DIMD + c0 + l] = acc[i];
}

// ---------------- edge-parallel scatter: agg[dst] += H[src] * norm ----------------

__global__ __launch_bounds__(256) void k_scatter(const float* __restrict__ H,
                                                 const int* __restrict__ src,
                                                 const int* __restrict__ dst,
                                                 const float* __restrict__ dis,
                                                 float* __restrict__ agg, int nE) {
    long long t = (long long)blockIdx.x * blockDim.x + threadIdx.x;
    int e = (int)(t >> 6);
    int f = (int)(t & 63);
    if (e < nE) {
        int s = src[e], d = dst[e];
        float nrm = dis[s] * dis[d];
        float v = H[(size_t)s * DIMD + f] * nrm;
        __hip_atomic_fetch_add(&agg[(size_t)d * DIMD + f], v, __ATOMIC_RELAXED,
                               __HIP_MEMORY_SCOPE_AGENT);
    }
}

// ---------------- epilogue: self-loop + bias + ReLU (in place) ----------------

__global__ __launch_bounds__(256) void k_fuse(const float* __restrict__ H,
                                              const float* __restrict__ invdeg,
                                              const float* __restrict__ b,
                                              float* __restrict__ agg, int n) {
    long long t = (long long)blockIdx.x * blockDim.x + threadIdx.x;
    int i = (int)(t >> 6);
    int f = (int)(t & 63);
    if (i < n) {
        float v = agg[t] + H[t] * invdeg[i] + b[f];
        agg[t] = v > 0.0f ? v : 0.0f;
    }
}

// ---------------- global_add_pool ----------------

__global__ __launch_bounds__(256) void k_pool(const float* __restrict__ h,
                                              const int* __restrict__ batch,
                                              float* __restrict__ pooled, int n) {
    long long t = (long long)blockIdx.x * blockDim.x + threadIdx.x;
    int i = (int)(t >> 6);
    int f = (int)(t & 63);
    if (i < n)
        __hip_atomic_fetch_add(&pooled[(size_t)batch[i] * DIMD + f], h[t],
                               __ATOMIC_RELAXED, __HIP_MEMORY_SCOPE_AGENT);
}

// ---------------- head: out[g] = pooled[g] . lin_w + lin_b ----------------

__global__ __launch_bounds__(64) void k_head(const float* __restrict__ pooled,
                                             const float* __restrict__ lw,
                                             const float* __restrict__ lb,
                                             float* __restrict__ out) {
    int g = threadIdx.x;
    if (g < NGRAPHS) {
        float acc = lb[0];
        #pragma unroll
        for (int f = 0; f < DIMD; ++f) acc += pooled[g * DIMD + f] * lw[f];
        out[g] = acc;
    }
}

extern "C" void kernel_launch(void* const* d_in, const int* in_sizes, int n_in,
                              void* d_out, int out_size, void* d_ws, size_t ws_size,
                              hipStream_t stream) {
    const float* x     = (const float*)d_in[0];
    const int*   ei    = (const int*)d_in[1];
    const int*   batch = (const int*)d_in[2];
    const float* W1 = (const float*)d_in[3];
    const float* b1 = (const float*)d_in[4];
    const float* W2 = (const float*)d_in[5];
    const float* b2 = (const float*)d_in[6];
    const float* W3 = (const float*)d_in[7];
    const float* b3 = (const float*)d_in[8];
    const float* lw = (const float*)d_in[9];
    const float* lb = (const float*)d_in[10];
    float* out = (float*)d_out;

    const int nE = in_sizes[1] / 2;   // edge_index is [2, E] flattened
    const int* src = ei;              // edge_index[0]
    const int* dst = ei + nE;         // edge_index[1]

    // Workspace layout (floats): H | bufA | deg | dis | invdeg | pooled  (~52.5 MB)
    float* ws     = (float*)d_ws;
    float* H      = ws;
    float* bufA   = H    + (size_t)NNODES * DIMD;
    float* deg    = bufA + (size_t)NNODES * DIMD;
    float* dis    = deg  + NNODES;
    float* invd   = dis  + NNODES;
    float* pooled = invd + NNODES;

    // --- symmetric normalization ---
    k_init_deg <<<(NNODES + 255) / 256, 256, 0, stream>>>(deg, NNODES);
    k_deg_count<<<(nE + 255) / 256,     256, 0, stream>>>(dst, deg, nE);
    k_dis      <<<(NNODES + 255) / 256, 256, 0, stream>>>(deg, dis, invd, NNODES);

    const float* Ws[3] = {W1, W2, W3};
    const float* bs[3] = {b1, b2, b3};
    const long long eThreads = (long long)nE * DIMD;
    const int eBlocks = (int)((eThreads + 255) / 256);
    const int nBlocks = (int)(((long long)NNODES * DIMD + 255) / 256);

    // Layer input dies after its GEMM, so two big buffers suffice:
    // GEMM reads `in` -> H; then `bufA` is zeroed, scattered into, fused in place.
    const float* in = x;
    for (int l = 0; l < 3; ++l) {
        k_gemm64<<<NNODES / 16, 128, 0, stream>>>(in, Ws[l], H);
        hipMemsetAsync(bufA, 0, (size_t)NNODES * DIMD * sizeof(float), stream);
        k_scatter<<<eBlocks, 256, 0, stream>>>(H, src, dst, dis, bufA, nE);
        k_fuse   <<<nBlocks, 256, 0, stream>>>(H, invd, bs[l], bufA, NNODES);
        in = bufA;
    }

    hipMemsetAsync(pooled, 0, (size_t)NGRAPHS * DIMD * sizeof(float), stream);
    k_pool<<<nBlocks, 256, 0, stream>>>(bufA, batch, pooled, NNODES);
    k_head<<<1, 64, 0, stream>>>(pooled, lw, lb, out);
}